// Nonlinear_1760936591660
// MI455X (gfx1250) — compile-verified
//
#include <hip/hip_runtime.h>

// Reference: out[:, :256] = f_{c%4}(in[:, :256])  with f = [id, sin, cos, sigmoid]
//            out[:, 256:] = in[:, 256::2] * in[:, 257::2]
//
// Pure streaming op: 384 MiB read + 256 MiB written => ~28 us floor @ 23.3 TB/s.
// Memory-bound; no contraction dim so WMMA does not apply. Strategy: two
// branch-free kernels, one thread per float4 of output, B128 non-temporal
// loads/stores (working set streams once and exceeds the 192 MB L2).

typedef __attribute__((ext_vector_type(4))) float v4f;

static constexpr int U_COLS  = 256;
static constexpr int IN_ROW  = 768;   // 256 unary + 512 binary inputs
static constexpr int OUT_ROW = 512;   // 256 unary + 256 binary outputs
// 64 float4 chunks per row in each half; 64 % 32 == 0 -> every wave32 convergent.

__device__ __forceinline__ float fast_sigmoid(float x) {
    // 1/(1+e^-x): v_exp_f32 + v_rcp_f32 on the TRANS units (free under memory bound)
    return __builtin_amdgcn_rcpf(1.0f + __expf(-x));
}

// One thread: 16B in -> 16B out. Aligned float4 maps component-wise to
// (identity, sin, cos, sigmoid) because the function pattern repeats every 4 cols.
__global__ __launch_bounds__(256) void unary_k(const float* __restrict__ in,
                                               float* __restrict__ out,
                                               int nChunks) {
    const int idx = (int)(blockIdx.x * blockDim.x + threadIdx.x);
    if (idx >= nChunks) return;
    const int row = idx >> 6;          // / 64 chunks per row
    const int c   = idx & 63;

    const v4f x = __builtin_nontemporal_load(
        (const v4f*)(in + (size_t)row * IN_ROW + (size_t)c * 4));
    v4f y;
    y.x = x.x;
    y.y = __sinf(x.y);
    y.z = __cosf(x.z);
    y.w = fast_sigmoid(x.w);
    __builtin_nontemporal_store(
        y, (v4f*)(out + (size_t)row * OUT_ROW + (size_t)c * 4));
}

// One thread: 32B in (two B128) -> 16B out of adjacent-pair products.
__global__ __launch_bounds__(256) void binary_k(const float* __restrict__ in,
                                                float* __restrict__ out,
                                                int nChunks) {
    const int idx = (int)(blockIdx.x * blockDim.x + threadIdx.x);
    if (idx >= nChunks) return;
    const int row = idx >> 6;
    const int c   = idx & 63;

    const v4f* p = (const v4f*)(in + (size_t)row * IN_ROW + U_COLS + (size_t)c * 8);
    const v4f a = __builtin_nontemporal_load(p);
    const v4f b = __builtin_nontemporal_load(p + 1);
    v4f y;
    y.x = a.x * a.y;
    y.y = a.z * a.w;
    y.z = b.x * b.y;
    y.w = b.z * b.w;
    __builtin_nontemporal_store(
        y, (v4f*)(out + (size_t)row * OUT_ROW + U_COLS + (size_t)c * 4));
}

extern "C" void kernel_launch(void* const* d_in, const int* in_sizes, int n_in,
                              void* d_out, int out_size, void* d_ws, size_t ws_size,
                              hipStream_t stream) {
    (void)n_in; (void)out_size; (void)d_ws; (void)ws_size;
    const float* in  = (const float*)d_in[0];
    float*       out = (float*)d_out;

    const int batch   = in_sizes[0] / IN_ROW;   // 131072
    const int nChunks = batch * 64;             // 8,388,608 float4 chunks per half

    const int threads = 256;                    // 8 wave32 per block
    const int blocks  = (nChunks + threads - 1) / threads;   // 32768, exact

    unary_k <<<blocks, threads, 0, stream>>>(in, out, nChunks);
    binary_k<<<blocks, threads, 0, stream>>>(in, out, nChunks);
}